// Encoder_46007689674960
// MI455X (gfx1250) — compile-verified
//
#include <hip/hip_runtime.h>
#include <hip/hip_bf16.h>

typedef __attribute__((ext_vector_type(2))) float v2f;
typedef __attribute__((ext_vector_type(8))) float v8f;

#define FEAT        512
#define EMB         256
#define NSAMP       10
#define BATCH       16384
#define BTILE       64
#define LDS_STRIDE  514   // 512 + 2-float pad -> conflict-free ds_load_b64
#define RRELU_SLOPE (11.0f / 48.0f)

// One workgroup: 256 threads (8 waves), computes full E=256 x BTILE=64 output tile.
__global__ __launch_bounds__(256)
void encoder_fused_kernel(const float* __restrict__ feat,
                          const float* __restrict__ weight,
                          const int*   __restrict__ nidx,
                          float*       __restrict__ out)
{
    __shared__ float smem[BTILE * LDS_STRIDE];   // 131,584 B of 320 KB WGP LDS

    const int tid = threadIdx.x;
    const int b0  = blockIdx.x * BTILE;

    // ---------------- Phase 1: gather + mean -> LDS agg[64][512] ----------------
    // Thread t covers features f = t and f = t + 256 (coalesced 1KB segments).
    for (int b = 0; b < BTILE; ++b) {
        const int* row = nidx + (size_t)(b0 + b) * NSAMP;  // uniform -> scalar loads
        float acc0 = 0.0f, acc1 = 0.0f;
        #pragma unroll
        for (int s = 0; s < NSAMP; ++s) {
            const float* fr = feat + (size_t)row[s] * FEAT;
            acc0 += fr[tid];
            acc1 += fr[tid + 256];
        }
        smem[b * LDS_STRIDE + tid]       = acc0 * 0.1f;
        smem[b * LDS_STRIDE + tid + 256] = acc1 * 0.1f;
    }
    __syncthreads();

    // ---------------- Phase 2: y = W(256x512) * agg^T(512x64) via WMMA f32 -----
    const int wave  = tid >> 5;      // 0..7 -> E rows [32*wave, 32*wave+32)
    const int lane  = tid & 31;
    const int lhalf = lane >> 4;     // 0/1 -> K sub-offset
    const int l15   = lane & 15;     // M (for A) / N (for B)

    v8f acc[2][4];
    #pragma unroll
    for (int m = 0; m < 2; ++m)
        #pragma unroll
        for (int j = 0; j < 4; ++j)
            acc[m][j] = (v8f){0.f, 0.f, 0.f, 0.f, 0.f, 0.f, 0.f, 0.f};

    // A fragment rows for this wave's two 16-row E tiles
    const float* wrow0 = weight + (size_t)(wave * 32 + l15) * FEAT;
    const float* wrow1 = wrow0 + 16 * FEAT;

    for (int k = 0; k < FEAT; k += 4) {
        const int kk = k + 2 * lhalf;

        // A 16x4: lane holds M=l15, K = kk,kk+1 in 2 VGPRs (8B global load, L2-hot)
        v2f a0 = *(const v2f*)(wrow0 + kk);
        v2f a1 = *(const v2f*)(wrow1 + kk);

        // B 4x16: lane holds column N = j*16 + l15, K = kk,kk+1 (8B LDS load)
        v2f bf[4];
        #pragma unroll
        for (int j = 0; j < 4; ++j)
            bf[j] = *(const v2f*)(&smem[(j * 16 + l15) * LDS_STRIDE + kk]);

        #pragma unroll
        for (int j = 0; j < 4; ++j) {
            acc[0][j] = __builtin_amdgcn_wmma_f32_16x16x4_f32(
                false, a0, false, bf[j], (short)0, acc[0][j], false, false);
            acc[1][j] = __builtin_amdgcn_wmma_f32_16x16x4_f32(
                false, a1, false, bf[j], (short)0, acc[1][j], false, false);
        }
    }

    // ---------------- Phase 3: RReLU(eval) + store [E, B] row-major -------------
    // C/D layout: VGPR r, lanes 0-15 -> M=r, lanes 16-31 -> M=r+8; N = l15.
    #pragma unroll
    for (int m = 0; m < 2; ++m) {
        #pragma unroll
        for (int j = 0; j < 4; ++j) {
            const int bcol = b0 + j * 16 + l15;
            #pragma unroll
            for (int r = 0; r < 8; ++r) {
                const int e = wave * 32 + m * 16 + 8 * lhalf + r;
                float v = acc[m][j][r];
                v = (v >= 0.0f) ? v : v * RRELU_SLOPE;
                out[(size_t)e * BATCH + bcol] = v;
            }
        }
    }
}

extern "C" void kernel_launch(void* const* d_in, const int* in_sizes, int n_in,
                              void* d_out, int out_size, void* d_ws, size_t ws_size,
                              hipStream_t stream) {
    const float* feat   = (const float*)d_in[0];   // [100000, 512]
    const float* weight = (const float*)d_in[1];   // [256, 512]
    const int*   nidx   = (const int*)d_in[2];     // [16384, 10]
    float*       out    = (float*)d_out;           // [256, 16384]

    dim3 grid(BATCH / BTILE);   // 256 workgroups
    dim3 block(256);            // 8 waves
    encoder_fused_kernel<<<grid, block, 0, stream>>>(feat, weight, nidx, out);
}